// SAE_78434692759816
// MI455X (gfx1250) — compile-verified
//
#include <hip/hip_runtime.h>
#include <hip/hip_bf16.h>

// ---------------------------------------------------------------------------
// Shapes (fixed by the reference)
// ---------------------------------------------------------------------------
#define B_  512
#define V_  50000
#define KDIM 50
#define EN_ 200
#define D_  384
#define S_  64
#define BN_EPS 1e-5f

typedef __attribute__((ext_vector_type(2))) float v2f;
typedef __attribute__((ext_vector_type(8))) float v8f;
typedef int v4i __attribute__((vector_size(4 * sizeof(int))));
typedef __attribute__((address_space(1))) v4i global_v4i;
typedef __attribute__((address_space(3))) v4i lds_v4i;

// ---------------------------------------------------------------------------
// CDNA5 async-copy (GLOBAL_LOAD_ASYNC_TO_LDS) support, guarded so that an
// older toolchain falls back to the proven synchronous staging path.
// Probe result (round 2): builtin exists; signature is
//   (v4i addrspace(1)* src, v4i addrspace(3)* dst, imm offset, imm cpol)
// ---------------------------------------------------------------------------
#if defined(__has_builtin)
#if __has_builtin(__builtin_amdgcn_global_load_async_to_lds_b128)
#define HAVE_ASYNC_LDS 1
#endif
#endif

#if defined(__has_builtin)
#if __has_builtin(__builtin_amdgcn_s_wait_asynccnt)
#define WAIT_ASYNC(n) __builtin_amdgcn_s_wait_asynccnt(n)
#endif
#endif
#ifndef WAIT_ASYNC
#define WAIT_ASYNC(n) asm volatile("s_wait_asynccnt %0" ::"i"(n) : "memory")
#endif

// ---------------------------------------------------------------------------
// WMMA fp32 GEMM:  C[M,N] = act( A[M,Ka] @ W[N,Ka]^T + bias[N] )
// One wave per block, 16x16 tile, double-buffered 64-wide K staging in LDS.
//   A (16x4):  lane<16 -> M=lane, v0=K(2*half), v1=K(2*half+1)
//   B (4x16):  lane&15 -> N,      v0=K(2*half), v1=K(2*half+1)
//   C/D:       vgpr r -> M = r + 8*half,  N = lane&15
// ---------------------------------------------------------------------------
#define KT   64
#define LSTR 68

__device__ __forceinline__ void stage_sync_tile(
    const float* __restrict__ A, int lda, const float* __restrict__ W, int ldw,
    int row0, int col0, int M, int N, int Ka, int k0,
    float* sAbuf, float* sBbuf, int lane)
{
    #pragma unroll
    for (int i = 0; i < (16 * KT) / 32; ++i) {
        int idx = lane + 32 * i;
        int r = idx >> 6;           // KT == 64
        int c = idx & 63;
        int gk = k0 + c;
        float av = 0.f, wv = 0.f;
        if (gk < Ka) {
            int ar = row0 + r;
            if (ar < M) av = A[(size_t)ar * lda + gk];
            int wr = col0 + r;
            if (wr < N) wv = W[(size_t)wr * ldw + gk];
        }
        sAbuf[r * LSTR + c] = av;
        sBbuf[r * LSTR + c] = wv;
    }
}

#if defined(HAVE_ASYNC_LDS)
// Issue 8 async B128 transfers per lane covering one 16xKT tile.
__device__ __forceinline__ void stage_async_tile(
    const float* __restrict__ G, int ld, int rowBase, int k0,
    float* lbuf, int lane)
{
    #pragma unroll
    for (int i = 0; i < 8; ++i) {
        int idx = lane + 32 * i;     // 256 float4 groups
        int r = idx >> 4;            // 16 groups per row
        int c4 = (idx & 15) * 4;
        const float* src = G + (size_t)(rowBase + r) * ld + k0 + c4;
        float* dst = lbuf + r * LSTR + c4;
        __builtin_amdgcn_global_load_async_to_lds_b128(
            (global_v4i*)src, (lds_v4i*)dst, 0, 0);
    }
}
#endif

__device__ __forceinline__ v8f wmma_tile_compute(const float* sAbuf, const float* sBbuf,
                                                 int lp, int half, v8f acc)
{
    #pragma unroll
    for (int kk = 0; kk < KT / 4; ++kk) {
        const int base = 4 * kk + 2 * half;
        v2f a, b;
        a.x = sAbuf[lp * LSTR + base];
        a.y = sAbuf[lp * LSTR + base + 1];
        b.x = sBbuf[lp * LSTR + base];
        b.y = sBbuf[lp * LSTR + base + 1];
        acc = __builtin_amdgcn_wmma_f32_16x16x4_f32(
            false, a, false, b, (short)0, acc, false, false);
    }
    return acc;
}

__global__ __launch_bounds__(32)
void wmma_gemm_bias_act(const float* __restrict__ A, int lda,
                        const float* __restrict__ W, int ldw,
                        const float* __restrict__ bias,
                        float* __restrict__ C, int ldc,
                        int M, int N, int Ka, int act)
{
    __shared__ float sA[2][16 * LSTR];
    __shared__ float sB[2][16 * LSTR];

    const int lane = threadIdx.x;
    const int half = lane >> 4;
    const int lp   = lane & 15;
    const int row0 = blockIdx.x * 16;
    const int col0 = blockIdx.y * 16;
    const int ntiles = (Ka + KT - 1) / KT;
    const bool edgeFull = (row0 + 16 <= M) && (col0 + 16 <= N);

    v8f acc = {0.f, 0.f, 0.f, 0.f, 0.f, 0.f, 0.f, 0.f};

#if defined(HAVE_ASYNC_LDS)
    bool async_cur = edgeFull && (KT <= Ka);
    if (async_cur) {
        stage_async_tile(A, lda, row0, 0, sA[0], lane);
        stage_async_tile(W, ldw, col0, 0, sB[0], lane);
    } else {
        stage_sync_tile(A, lda, W, ldw, row0, col0, M, N, Ka, 0, sA[0], sB[0], lane);
    }
    for (int t = 0; t < ntiles; ++t) {
        const int buf = t & 1;
        bool async_next = false;
        if (t + 1 < ntiles) {
            const int k1 = (t + 1) * KT;
            async_next = edgeFull && (k1 + KT <= Ka);
            if (async_next) {
                stage_async_tile(A, lda, row0, k1, sA[buf ^ 1], lane);
                stage_async_tile(W, ldw, col0, k1, sB[buf ^ 1], lane);
            } else {
                stage_sync_tile(A, lda, W, ldw, row0, col0, M, N, Ka, k1,
                                sA[buf ^ 1], sB[buf ^ 1], lane);
            }
        }
        if (async_cur) { if (async_next) WAIT_ASYNC(16); else WAIT_ASYNC(0); }
        __syncthreads();
        acc = wmma_tile_compute(sA[buf], sB[buf], lp, half, acc);
        __syncthreads();
        async_cur = async_next;
    }
#else
    for (int t = 0; t < ntiles; ++t) {
        const int k0 = t * KT;
        if (k0 + KT < Ka) {
            if (row0 + lp < M)
                __builtin_prefetch(A + (size_t)(row0 + lp) * lda + k0 + KT, 0, 3);
            if (col0 + lp < N)
                __builtin_prefetch(W + (size_t)(col0 + lp) * ldw + k0 + KT, 0, 3);
        }
        stage_sync_tile(A, lda, W, ldw, row0, col0, M, N, Ka, k0, sA[0], sB[0], lane);
        __syncthreads();
        acc = wmma_tile_compute(sA[0], sB[0], lp, half, acc);
        __syncthreads();
    }
#endif

    const int n = col0 + lp;
    if (n < N) {
        const float bval = bias ? bias[n] : 0.f;
        #pragma unroll
        for (int r = 0; r < 8; ++r) {
            int m = row0 + r + 8 * half;
            if (m < M) {
                float v = acc[r] + bval;
                if (act == 1)
                    v = fmaxf(v, 0.f) + log1pf(expf(-fabsf(v)));
                C[(size_t)m * ldc + n] = v;
            }
        }
    }
}

// ---------------------------------------------------------------------------
// Attention: per-document masked softmax attention over S=64 sub-embeddings.
// ---------------------------------------------------------------------------
__global__ __launch_bounds__(128)
void attn_kernel(const float* __restrict__ subs, const float* __restrict__ q,
                 const int* __restrict__ lengths, float* __restrict__ dvec)
{
    const int b = blockIdx.x;
    const int t = threadIdx.x;
    __shared__ float sq[D_];
    __shared__ float salpha[S_];
    __shared__ float red[2];

    const float* qb = q + (size_t)b * D_;
    const float* sb = subs + (size_t)b * S_ * D_;
    for (int j = t; j < D_; j += 128) sq[j] = qb[j];
    __syncthreads();

    const int len = lengths[b];
    if (t < S_) {
        float s = 0.f;
        const float* row = sb + (size_t)t * D_;
        for (int j = 0; j < D_; ++j) s += row[j] * sq[j];
        s *= (1.0f / sqrtf((float)D_));
        salpha[t] = (t < len) ? s : -1e30f;
    }
    __syncthreads();
    if (t == 0) {
        float m = -__builtin_inff();
        for (int s = 0; s < S_; ++s) m = fmaxf(m, salpha[s]);
        float sum = 0.f;
        for (int s = 0; s < S_; ++s) sum += expf(salpha[s] - m);
        red[0] = m; red[1] = sum;
    }
    __syncthreads();
    if (t < S_) salpha[t] = expf(salpha[t] - red[0]) / red[1];
    __syncthreads();

    float* db = dvec + (size_t)b * D_;
    for (int j = t; j < D_; j += 128) {
        float accv = 0.f;
        for (int s = 0; s < S_; ++s) accv += salpha[s] * sb[(size_t)s * D_ + j];
        db[j] = (len > 0) ? accv : sq[j];
    }
}

// ---------------------------------------------------------------------------
// Column-wise batch stats (training-mode BatchNorm, biased variance).
// ---------------------------------------------------------------------------
__global__ __launch_bounds__(64)
void bn_stats(const float* __restrict__ src, int stride, int n, int rows,
              float* __restrict__ mean, float* __restrict__ rstd)
{
    const int k = threadIdx.x;
    if (k >= n) return;
    float s = 0.f, s2 = 0.f;
    for (int b = 0; b < rows; ++b) {
        float v = src[(size_t)b * stride + k];
        s += v; s2 += v * v;
    }
    float m = s / rows;
    float var = s2 / rows - m * m;
    mean[k] = m;
    rstd[k] = rsqrtf(var + BN_EPS);
}

// ---------------------------------------------------------------------------
// Per-row: BN-apply, reparameterize, theta = softmax(z), per-row KLD.
// theta stored with padded stride 64; pad columns zeroed (decoder reads K=52).
// ---------------------------------------------------------------------------
__global__ __launch_bounds__(64)
void ztheta_kernel(const float* __restrict__ mu_pre, const float* __restrict__ lv_pre,
                   const float* __restrict__ mean_mu, const float* __restrict__ rstd_mu,
                   const float* __restrict__ mean_lv, const float* __restrict__ rstd_lv,
                   const float* __restrict__ bn_mean_b, const float* __restrict__ bn_logvar_b,
                   const float* __restrict__ eps, const float* __restrict__ mu2,
                   const float* __restrict__ var2,
                   float* __restrict__ theta, float* __restrict__ kld)
{
    const int b = blockIdx.x;
    const int k = threadIdx.x;   // 0..63
    __shared__ float sz[64];
    __shared__ float skld[64];
    __shared__ float red[2];

    float z = -__builtin_inff(), kterm = 0.f;
    if (k < KDIM) {
        float mu = (mu_pre[b * 64 + k] - mean_mu[k]) * rstd_mu[k] + bn_mean_b[k];
        float lv = (lv_pre[b * 64 + k] - mean_lv[k]) * rstd_lv[k] + bn_logvar_b[k];
        z = mu + eps[b * KDIM + k] * expf(0.5f * lv);
        float var = expf(lv);
        float dm = mu - mu2[k];
        kterm = var / var2[k] + dm * dm / var2[k] + logf(var2[k]) - lv;
    }
    sz[k] = z; skld[k] = kterm;
    __syncthreads();
    if (k == 0) {
        float m = -__builtin_inff();
        for (int i = 0; i < KDIM; ++i) m = fmaxf(m, sz[i]);
        float s = 0.f, ks = 0.f;
        for (int i = 0; i < KDIM; ++i) { s += expf(sz[i] - m); ks += skld[i]; }
        red[0] = m; red[1] = s;
        kld[b] = 0.5f * (ks - (float)KDIM);
    }
    __syncthreads();
    if (k < KDIM) theta[b * 64 + k] = expf(z - red[0]) / red[1];
    else          theta[b * 64 + k] = 0.f;
}

// ---------------------------------------------------------------------------
// G = theta^T theta (KxK) and theta_mean (K).
// ---------------------------------------------------------------------------
__global__ __launch_bounds__(256)
void theta_stats(const float* __restrict__ theta,
                 float* __restrict__ G, float* __restrict__ tmean)
{
    const int t = threadIdx.x;
    for (int e = t; e < KDIM * KDIM; e += 256) {
        int i = e / KDIM, j = e % KDIM;
        float s = 0.f;
        for (int b = 0; b < B_; ++b) s += theta[b * 64 + i] * theta[b * 64 + j];
        G[e] = s;
    }
    if (t < KDIM) {
        float s = 0.f;
        for (int b = 0; b < B_; ++b) s += theta[b * 64 + t];
        tmean[t] = s / (float)B_;
    }
}

// ---------------------------------------------------------------------------
// Decoder BN column stats without materializing h = theta @ Wd^T:
//   mean_v = tmean . wd_v ;  E[h^2]_v = (wd_v^T G wd_v)/B
// ---------------------------------------------------------------------------
__global__ __launch_bounds__(256)
void dec_stats(const float* __restrict__ Wd, const float* __restrict__ G,
               const float* __restrict__ tmean,
               float* __restrict__ meanv, float* __restrict__ rstdv)
{
    __shared__ float sG[KDIM * KDIM];
    __shared__ float stm[KDIM];
    const int t = threadIdx.x;
    for (int e = t; e < KDIM * KDIM; e += 256) sG[e] = G[e];
    if (t < KDIM) stm[t] = tmean[t];
    __syncthreads();

    const int v = blockIdx.x * 256 + t;
    if (v >= V_) return;
    const float* wd = Wd + (size_t)v * KDIM;
    float w[KDIM];
    #pragma unroll
    for (int k = 0; k < KDIM; ++k) w[k] = wd[k];
    float mean = 0.f;
    #pragma unroll
    for (int k = 0; k < KDIM; ++k) mean += stm[k] * w[k];
    float quad = 0.f;
    for (int i = 0; i < KDIM; ++i) {
        float gi = 0.f;
        for (int j = 0; j < KDIM; ++j) gi += sG[i * KDIM + j] * w[j];
        quad += w[i] * gi;
    }
    float var = quad / (float)B_ - mean * mean;
    meanv[v] = mean;
    rstdv[v] = rsqrtf(var + BN_EPS);
}

// ---------------------------------------------------------------------------
// WMMA decoder sweeps. Each block: 1 wave, 16 batch rows x VB_COLS vocab cols.
// h tile (16x16) via 13 x V_WMMA_F32_16X16X4_F32 (K = 50, zero-padded to 52).
// C layout: vgpr r -> batch row (r + 8*half), lane&15 -> vocab col.
// ---------------------------------------------------------------------------
#define VB_COLS 1024
#define NVB ((V_ + VB_COLS - 1) / VB_COLS)   // 49
#define KSTEP 13                              // ceil(50/4)

__global__ __launch_bounds__(32)
void dec_lse_wmma(const float* __restrict__ theta, const float* __restrict__ Wd,
                  const float* __restrict__ meanv, const float* __restrict__ rstdv,
                  const float* __restrict__ bn_dec_b,
                  float* __restrict__ Pm, float* __restrict__ Ps)
{
    const int lane = threadIdx.x, half = lane >> 4, lp = lane & 15;
    const int rb = blockIdx.x;                 // batch-row block (16 rows)
    const int vb = blockIdx.y;                 // vocab block
    const int b0 = rb * 16;
    const int vstart = vb * VB_COLS;
    const int vend = (vstart + VB_COLS < V_) ? (vstart + VB_COLS) : V_;

    // Loop-invariant A fragments (theta rows; padded zeros above K=50)
    v2f afr[KSTEP];
    #pragma unroll
    for (int kk = 0; kk < KSTEP; ++kk) {
        int c = 4 * kk + 2 * half;
        afr[kk].x = theta[(b0 + lp) * 64 + c];
        afr[kk].y = theta[(b0 + lp) * 64 + c + 1];
    }

    float om[8], os[8];
    #pragma unroll
    for (int r = 0; r < 8; ++r) { om[r] = -__builtin_inff(); os[r] = 0.f; }

    for (int v0 = vstart; v0 < vend; v0 += 16) {       // V_ % 16 == 0
        const int v = v0 + lp;
        if (v0 + 16 < vend)
            __builtin_prefetch(Wd + (size_t)(v + 16) * KDIM, 0, 3);
        const float mv = meanv[v];
        const float rv = rstdv[v];
        const float bb = bn_dec_b[v];
        const float* wrow = Wd + (size_t)v * KDIM;

        v8f acc = {0.f, 0.f, 0.f, 0.f, 0.f, 0.f, 0.f, 0.f};
        #pragma unroll
        for (int kk = 0; kk < KSTEP; ++kk) {
            int c = 4 * kk + 2 * half;
            v2f b;
            b.x = (c     < KDIM) ? wrow[c]     : 0.f;
            b.y = (c + 1 < KDIM) ? wrow[c + 1] : 0.f;
            acc = __builtin_amdgcn_wmma_f32_16x16x4_f32(
                false, afr[kk], false, b, (short)0, acc, false, false);
        }
        #pragma unroll
        for (int r = 0; r < 8; ++r) {
            float g = (acc[r] - mv) * rv + bb;
            if (g > om[r]) { os[r] = os[r] * expf(om[r] - g) + 1.f; om[r] = g; }
            else           { os[r] += expf(g - om[r]); }
        }
    }

    // Merge (m,s) across the 16 lanes holding the same batch row
    #pragma unroll
    for (int r = 0; r < 8; ++r) {
        float m = om[r], s = os[r];
        #pragma unroll
        for (int d = 1; d < 16; d <<= 1) {
            float m2 = __shfl_xor(m, d, 16);
            float s2 = __shfl_xor(s, d, 16);
            float mm = fmaxf(m, m2);
            s = s * expf(m - mm) + s2 * expf(m2 - mm);
            m = mm;
        }
        om[r] = m; os[r] = s;
    }
    if (lp == 0) {
        #pragma unroll
        for (int r = 0; r < 8; ++r) {
            int idx = (rb * NVB + vb) * 16 + 8 * half + r;
            Pm[idx] = om[r];
            Ps[idx] = os[r];
        }
    }
}

__global__ __launch_bounds__(512)
void dec_lse_combine(const float* __restrict__ Pm, const float* __restrict__ Ps,
                     float* __restrict__ lse)
{
    const int b = threadIdx.x;                 // 512 threads == B_
    const int rb = b >> 4, ri = b & 15;
    float m = -__builtin_inff(), s = 0.f;
    for (int vb = 0; vb < NVB; ++vb) {
        const int idx = (rb * NVB + vb) * 16 + ri;
        float m2 = Pm[idx], s2 = Ps[idx];
        float mm = fmaxf(m, m2);
        s = s * expf(m - mm) + s2 * expf(m2 - mm);
        m = mm;
    }
    lse[b] = m + logf(s);
}

__global__ __launch_bounds__(32)
void dec_loss_wmma(const float* __restrict__ theta, const float* __restrict__ Wd,
                   const float* __restrict__ meanv, const float* __restrict__ rstdv,
                   const float* __restrict__ bn_dec_b, const float* __restrict__ lse,
                   const float* __restrict__ x, float* __restrict__ P2)
{
    const int lane = threadIdx.x, half = lane >> 4, lp = lane & 15;
    const int rb = blockIdx.x;
    const int vb = blockIdx.y;
    const int b0 = rb * 16;
    const int vstart = vb * VB_COLS;
    const int vend = (vstart + VB_COLS < V_) ? (vstart + VB_COLS) : V_;

    v2f afr[KSTEP];
    #pragma unroll
    for (int kk = 0; kk < KSTEP; ++kk) {
        int c = 4 * kk + 2 * half;
        afr[kk].x = theta[(b0 + lp) * 64 + c];
        afr[kk].y = theta[(b0 + lp) * 64 + c + 1];
    }
    float lse_s[8];
    #pragma unroll
    for (int r = 0; r < 8; ++r) lse_s[r] = lse[b0 + 8 * half + r];

    float accs[8];
    #pragma unroll
    for (int r = 0; r < 8; ++r) accs[r] = 0.f;

    for (int v0 = vstart; v0 < vend; v0 += 16) {
        const int v = v0 + lp;
        if (v0 + 16 < vend)
            __builtin_prefetch(Wd + (size_t)(v + 16) * KDIM, 0, 3);
        const float mv = meanv[v];
        const float rv = rstdv[v];
        const float bb = bn_dec_b[v];
        const float* wrow = Wd + (size_t)v * KDIM;

        v8f acc = {0.f, 0.f, 0.f, 0.f, 0.f, 0.f, 0.f, 0.f};
        #pragma unroll
        for (int kk = 0; kk < KSTEP; ++kk) {
            int c = 4 * kk + 2 * half;
            v2f b;
            b.x = (c     < KDIM) ? wrow[c]     : 0.f;
            b.y = (c + 1 < KDIM) ? wrow[c + 1] : 0.f;
            acc = __builtin_amdgcn_wmma_f32_16x16x4_f32(
                false, afr[kk], false, b, (short)0, acc, false, false);
        }
        #pragma unroll
        for (int r = 0; r < 8; ++r) {
            float g = (acc[r] - mv) * rv + bb;
            float xv = x[(size_t)(b0 + 8 * half + r) * V_ + v];
            accs[r] += xv * logf(expf(g - lse_s[r]) + 1e-10f);
        }
    }

    #pragma unroll
    for (int r = 0; r < 8; ++r) {
        float s = accs[r];
        #pragma unroll
        for (int d = 1; d < 16; d <<= 1) s += __shfl_xor(s, d, 16);
        accs[r] = s;
    }
    if (lp == 0) {
        #pragma unroll
        for (int r = 0; r < 8; ++r)
            P2[(rb * NVB + vb) * 16 + 8 * half + r] = accs[r];
    }
}

__global__ __launch_bounds__(512)
void final_loss(const float* __restrict__ P2, const float* __restrict__ kld,
                float* __restrict__ out)
{
    __shared__ float red[512];
    const int b = threadIdx.x;                 // 512 threads == B_
    const int rb = b >> 4, ri = b & 15;
    float s = 0.f;
    for (int vb = 0; vb < NVB; ++vb) s += P2[(rb * NVB + vb) * 16 + ri];
    red[b] = -s + kld[b];
    __syncthreads();
    for (int off = 256; off > 0; off >>= 1) {
        if (b < off) red[b] += red[b + off];
        __syncthreads();
    }
    if (b == 0) out[0] = red[0] / (float)B_;
}

// ---------------------------------------------------------------------------
// Host-side orchestration
// ---------------------------------------------------------------------------
extern "C" void kernel_launch(void* const* d_in, const int* in_sizes, int n_in,
                              void* d_out, int out_size, void* d_ws, size_t ws_size,
                              hipStream_t stream)
{
    const float* x          = (const float*)d_in[0];   // (B,V)
    const float* subs       = (const float*)d_in[1];   // (B,S,D)
    const float* eps        = (const float*)d_in[2];   // (B,K)
    const int*   lengths    = (const int*)d_in[3];     // (B,)
    const float* W11        = (const float*)d_in[4];   // (EN,V)
    const float* b11        = (const float*)d_in[5];
    const float* W12        = (const float*)d_in[6];   // (D,EN)
    const float* b12        = (const float*)d_in[7];
    const float* W21        = (const float*)d_in[8];   // (K,D)
    const float* b21        = (const float*)d_in[9];
    const float* W22        = (const float*)d_in[10];  // (K,D)
    const float* b22        = (const float*)d_in[11];
    const float* Wd         = (const float*)d_in[12];  // (V,K)
    const float* bn_mean_b  = (const float*)d_in[13];
    const float* bn_logvar_b= (const float*)d_in[14];
    const float* bn_dec_b   = (const float*)d_in[15];  // (V,)
    const float* mu2        = (const float*)d_in[16];  // (1,K)
    const float* var2       = (const float*)d_in[17];  // (1,K)
    float* out = (float*)d_out;

    float* ws = (float*)d_ws;
    size_t off = 0;
    float* e1      = ws + off; off += (size_t)B_ * EN_;
    float* q       = ws + off; off += (size_t)B_ * D_;
    float* dvec    = ws + off; off += (size_t)B_ * D_;
    float* mu_pre  = ws + off; off += (size_t)B_ * 64;
    float* lv_pre  = ws + off; off += (size_t)B_ * 64;
    float* mean_mu = ws + off; off += 64;
    float* rstd_mu = ws + off; off += 64;
    float* mean_lv = ws + off; off += 64;
    float* rstd_lv = ws + off; off += 64;
    float* theta   = ws + off; off += (size_t)B_ * 64;
    float* kld     = ws + off; off += B_;
    float* G       = ws + off; off += KDIM * KDIM;
    float* tmean   = ws + off; off += 64;
    float* meanv   = ws + off; off += V_;
    float* rstdv   = ws + off; off += V_;
    float* Pm      = ws + off; off += (size_t)(B_ / 16) * NVB * 16;
    float* Ps      = ws + off; off += (size_t)(B_ / 16) * NVB * 16;
    float* P2      = ws + off; off += (size_t)(B_ / 16) * NVB * 16;
    float* lse     = ws + off; off += B_;
    (void)ws_size; (void)n_in; (void)in_sizes; (void)out_size;

    // 1) e1 = softplus(x @ W11^T + b11)
    wmma_gemm_bias_act<<<dim3(B_ / 16, (EN_ + 15) / 16), 32, 0, stream>>>(
        x, V_, W11, V_, b11, e1, EN_, B_, EN_, V_, 1);

    // 2) q = softplus(e1 @ W12^T + b12)
    wmma_gemm_bias_act<<<dim3(B_ / 16, D_ / 16), 32, 0, stream>>>(
        e1, EN_, W12, EN_, b12, q, D_, B_, D_, EN_, 1);

    // 3) attention -> dvec
    attn_kernel<<<B_, 128, 0, stream>>>(subs, q, lengths, dvec);

    // 4) variational heads (padded stride 64)
    wmma_gemm_bias_act<<<dim3(B_ / 16, (KDIM + 15) / 16), 32, 0, stream>>>(
        dvec, D_, W21, D_, b21, mu_pre, 64, B_, KDIM, D_, 0);
    wmma_gemm_bias_act<<<dim3(B_ / 16, (KDIM + 15) / 16), 32, 0, stream>>>(
        dvec, D_, W22, D_, b22, lv_pre, 64, B_, KDIM, D_, 0);

    // 5) BN batch stats
    bn_stats<<<1, 64, 0, stream>>>(mu_pre, 64, KDIM, B_, mean_mu, rstd_mu);
    bn_stats<<<1, 64, 0, stream>>>(lv_pre, 64, KDIM, B_, mean_lv, rstd_lv);

    // 6) z, theta, per-row KLD
    ztheta_kernel<<<B_, 64, 0, stream>>>(mu_pre, lv_pre, mean_mu, rstd_mu,
                                         mean_lv, rstd_lv, bn_mean_b, bn_logvar_b,
                                         eps, mu2, var2, theta, kld);

    // 7) G = theta^T theta, theta_mean
    theta_stats<<<1, 256, 0, stream>>>(theta, G, tmean);

    // 8) decoder BN column stats
    dec_stats<<<(V_ + 255) / 256, 256, 0, stream>>>(Wd, G, tmean, meanv, rstdv);

    // 9) WMMA logsumexp sweep + combine
    dec_lse_wmma<<<dim3(B_ / 16, NVB), 32, 0, stream>>>(theta, Wd, meanv, rstdv,
                                                        bn_dec_b, Pm, Ps);
    dec_lse_combine<<<1, 512, 0, stream>>>(Pm, Ps, lse);

    // 10) WMMA loss sweep + final mean
    dec_loss_wmma<<<dim3(B_ / 16, NVB), 32, 0, stream>>>(theta, Wd, meanv, rstdv,
                                                         bn_dec_b, lse, x, P2);
    final_loss<<<1, 512, 0, stream>>>(P2, kld, out);
}